// time_BiFormerBlock_43138651521514
// MI455X (gfx1250) — compile-verified
//
#include <hip/hip_runtime.h>
#include <hip/hip_bf16.h>

// ---------------------------------------------------------------------------
// BiFormer block for MI455X (gfx1250, wave32).
// All channel-mixing matmuls use V_WMMA_F32_16X16X4_F32 (fp32 WMMA).
// ---------------------------------------------------------------------------

typedef float v2f __attribute__((ext_vector_type(2)));
typedef float v8f __attribute__((ext_vector_type(8)));

#define B_   16
#define C_   64
#define T_   2048
#define NR_  1024
#define H_   4
#define HD_  16
#define TK_  40
#define C2_  128

// 1/sqrt(1 + 1e-5)
#define RSQ  0.99999500003749968f
// 1/sqrt(2048)  (ortho FFT normalization, applied once per direction)
#define INVS 0.02209708691207961f

__device__ __forceinline__ v8f wmma4(v2f a, v2f b, v8f c) {
  // D = A(16x4 f32) * B(4x16 f32) + C(16x16 f32)
  return __builtin_amdgcn_wmma_f32_16x16x4_f32(false, a, false, b, (short)0, c,
                                               false, false);
}

// ---------------------------------------------------------------------------
// K1: xn = bn(x, n1_g, n1_b)   (B,C,T) elementwise
// ---------------------------------------------------------------------------
__global__ __launch_bounds__(256) void k_bn1(const float* __restrict__ x,
                                             const float* __restrict__ g,
                                             const float* __restrict__ bb,
                                             float* __restrict__ xn) {
  int i = blockIdx.x * 256 + threadIdx.x;          // 0 .. B*C*T-1
  int c = (i >> 11) & 63;
  xn[i] = x[i] * (g[c] * RSQ) + bb[c];
}

// ---------------------------------------------------------------------------
// K2: q = bn(conv3(xn,wq)), k = bn(conv3(xn,wk)), v = conv1(xn,wv)
// Output layout (B, T, C) for coalesced attention gathers.
// WMMA fp32: A = weight tile 16x4, B = activation tile 4x16.
// ---------------------------------------------------------------------------
__global__ __launch_bounds__(256) void k_qkv(
    const float* __restrict__ xn,
    const float* __restrict__ wq, const float* __restrict__ wk,
    const float* __restrict__ wv,
    const float* __restrict__ bqg, const float* __restrict__ bqb,
    const float* __restrict__ bkg, const float* __restrict__ bkb,
    float* __restrict__ q_t, float* __restrict__ k_t, float* __restrict__ v_t) {
  __shared__ float xs[64][34];                      // halo tile t0-1 .. t0+32
  int bid = blockIdx.x;
  int b   = bid >> 6;
  int t0  = (bid & 63) << 5;
  for (int i = threadIdx.x; i < 64 * 34; i += 256) {
    int ci = i / 34, tt = i % 34;
    int t  = t0 - 1 + tt;
    xs[ci][tt] = (t >= 0 && t < T_) ? xn[((size_t)b * C_ + ci) * T_ + t] : 0.f;
  }
  __syncthreads();
  int wave = threadIdx.x >> 5, lane = threadIdx.x & 31;
  int row = lane & 15, kh = lane >> 4;
  int co0 = (wave & 3) << 4, tn0 = (wave >> 2) << 4;
  int col = tn0 + row;
  v8f aq = {}, ak = {}, av = {};
  for (int c0 = 0; c0 < 64; c0 += 4) {
    int kA = c0 + (kh << 1);                        // K indices kA, kA+1
    v2f b0, b1, b2, a;
    b0.x = xs[kA][col];     b0.y = xs[kA + 1][col];
    b1.x = xs[kA][col + 1]; b1.y = xs[kA + 1][col + 1];
    b2.x = xs[kA][col + 2]; b2.y = xs[kA + 1][col + 2];
    const float* wqp = wq + (size_t)(co0 + row) * 192 + kA * 3;
    const float* wkp = wk + (size_t)(co0 + row) * 192 + kA * 3;
    const float* wvp = wv + (size_t)(co0 + row) * 64 + kA;
    a.x = wqp[0]; a.y = wqp[3]; aq = wmma4(a, b0, aq);
    a.x = wqp[1]; a.y = wqp[4]; aq = wmma4(a, b1, aq);
    a.x = wqp[2]; a.y = wqp[5]; aq = wmma4(a, b2, aq);
    a.x = wkp[0]; a.y = wkp[3]; ak = wmma4(a, b0, ak);
    a.x = wkp[1]; a.y = wkp[4]; ak = wmma4(a, b1, ak);
    a.x = wkp[2]; a.y = wkp[5]; ak = wmma4(a, b2, ak);
    a.x = wvp[0]; a.y = wvp[1]; av = wmma4(a, b1, av);
  }
#pragma unroll
  for (int v = 0; v < 8; v++) {
    int co = co0 + v + (kh << 3);
    int t  = t0 + tn0 + row;
    size_t o = ((size_t)b * T_ + t) * C_ + co;
    q_t[o] = aq[v] * (bqg[co] * RSQ) + bqb[co];
    k_t[o] = ak[v] * (bkg[co] * RSQ) + bkb[co];
    v_t[o] = av[v];
  }
}

// ---------------------------------------------------------------------------
// K3: q_r/k_r block means over RS=2, layout (B, NR, C)
// ---------------------------------------------------------------------------
__global__ __launch_bounds__(256) void k_reduce(const float* __restrict__ q_t,
                                                const float* __restrict__ k_t,
                                                float* __restrict__ qr,
                                                float* __restrict__ kr) {
  int i = blockIdx.x * 256 + threadIdx.x;          // 0 .. B*NR*C-1
  int c = i & 63;
  int r = (i >> 6) & 1023;
  int b = i >> 16;
  size_t s0 = ((size_t)b * T_ + 2 * r) * C_ + c;
  qr[i] = 0.5f * (q_t[s0] + q_t[s0 + C_]);
  kr[i] = 0.5f * (k_t[s0] + k_t[s0 + C_]);
}

// ---------------------------------------------------------------------------
// K4: routing scores (16 rows x 1024 cols per WG, WMMA fp32) + per-row top-40.
// Softmax-weighted sum downstream is permutation-invariant, so selection
// order does not matter.
// ---------------------------------------------------------------------------
__global__ __launch_bounds__(128) void k_route(const float* __restrict__ qr,
                                               const float* __restrict__ kr,
                                               int* __restrict__ idxo) {
  __shared__ float sc[16 * 1024];                  // 64 KB LDS
  int b  = blockIdx.x >> 6;
  int r0 = (blockIdx.x & 63) << 4;
  int wave = threadIdx.x >> 5, lane = threadIdx.x & 31;
  int row = lane & 15, kh = lane >> 4;
  for (int nt = wave; nt < 64; nt += 4) {
    int s0 = nt << 4;
    v8f acc = {};
    for (int c0 = 0; c0 < 64; c0 += 4) {
      int kA = c0 + (kh << 1);
      v2f a, bvec;
      const float* qp = qr + ((size_t)b * NR_ + (r0 + row)) * C_ + kA;
      const float* kp = kr + ((size_t)b * NR_ + (s0 + row)) * C_ + kA;
      a.x = qp[0]; a.y = qp[1];
      bvec.x = kp[0]; bvec.y = kp[1];
      acc = wmma4(a, bvec, acc);
    }
#pragma unroll
    for (int v = 0; v < 8; v++)
      sc[(v + (kh << 3)) * 1024 + s0 + row] = acc[v];
  }
  __syncthreads();
  // iterative top-40 per row; each lane keeps 32 scores in registers
  for (int rr = wave; rr < 16; rr += 4) {
    float vals[32];
    int base = rr * 1024 + lane;
#pragma unroll
    for (int j = 0; j < 32; j++) vals[j] = sc[base + j * 32];
    int* op = idxo + ((size_t)b * NR_ + (r0 + rr)) * TK_;
    for (int it = 0; it < TK_; it++) {
      float mv = -3.0e38f; int mj = 0;
#pragma unroll
      for (int j = 0; j < 32; j++)
        if (vals[j] > mv) { mv = vals[j]; mj = j; }
      int midx = mj * 32 + lane;
      for (int off = 16; off; off >>= 1) {
        float ov = __shfl_xor(mv, off, 32);
        int   oi = __shfl_xor(midx, off, 32);
        if (ov > mv || (ov == mv && oi < midx)) { mv = ov; midx = oi; }
      }
      if (lane == 0) op[it] = midx;
      int owner = midx & 31, kj = midx >> 5;
      if (lane == owner) {
#pragma unroll
        for (int j = 0; j < 32; j++)
          if (j == kj) vals[j] = -3.0e38f;
      }
    }
  }
}

// ---------------------------------------------------------------------------
// K5: gathered attention. One wave per (b,h,r): q(2x16) . Kg(80x16)^T,
// softmax, @ Vg(80x16). Gathers are 64B-coalesced from (B,T,C) layout.
// ---------------------------------------------------------------------------
__global__ __launch_bounds__(256) void k_attn(const float* __restrict__ q_t,
                                              const float* __restrict__ k_t,
                                              const float* __restrict__ v_t,
                                              const int* __restrict__ idx,
                                              float* __restrict__ o_t) {
  int wid  = blockIdx.x * 8 + (threadIdx.x >> 5);  // 0 .. B*H*NR-1
  int lane = threadIdx.x & 31;
  int r = wid & (NR_ - 1);
  int h = (wid >> 10) & (H_ - 1);
  int b = wid >> 12;
  const float* qp = q_t + ((size_t)b * T_ + 2 * r) * C_ + h * HD_;
  float q0[16], q1[16];
#pragma unroll
  for (int j = 0; j < 4; j++) {
    float4 f0 = *(const float4*)(qp + 4 * j);
    float4 f1 = *(const float4*)(qp + C_ + 4 * j);
    q0[4 * j] = f0.x; q0[4 * j + 1] = f0.y; q0[4 * j + 2] = f0.z; q0[4 * j + 3] = f0.w;
    q1[4 * j] = f1.x; q1[4 * j + 1] = f1.y; q1[4 * j + 2] = f1.z; q1[4 * j + 3] = f1.w;
  }
  const int* ip = idx + ((size_t)b * NR_ + r) * TK_;
  float l0[3], l1[3];
  int tkk[3];
#pragma unroll
  for (int sl = 0; sl < 3; sl++) {
    int kk = lane + 32 * sl;
    bool act = kk < 80;
    float d0 = 0.f, d1 = 0.f;
    int t = 0;
    if (act) {
      int blk = ip[kk >> 1];
      t = blk * 2 + (kk & 1);
      const float* kp = k_t + ((size_t)b * T_ + t) * C_ + h * HD_;
#pragma unroll
      for (int j = 0; j < 4; j++) {
        float4 kv = *(const float4*)(kp + 4 * j);
        d0 += q0[4*j]*kv.x + q0[4*j+1]*kv.y + q0[4*j+2]*kv.z + q0[4*j+3]*kv.w;
        d1 += q1[4*j]*kv.x + q1[4*j+1]*kv.y + q1[4*j+2]*kv.z + q1[4*j+3]*kv.w;
      }
    }
    tkk[sl] = t;
    l0[sl] = act ? d0 * 0.125f : -3.0e38f;          // SCALE = 64^-0.5
    l1[sl] = act ? d1 * 0.125f : -3.0e38f;
  }
  float m0 = fmaxf(fmaxf(l0[0], l0[1]), l0[2]);
  float m1 = fmaxf(fmaxf(l1[0], l1[1]), l1[2]);
  for (int off = 16; off; off >>= 1) {
    m0 = fmaxf(m0, __shfl_xor(m0, off, 32));
    m1 = fmaxf(m1, __shfl_xor(m1, off, 32));
  }
  float e0[3], e1[3], s0 = 0.f, s1 = 0.f;
#pragma unroll
  for (int sl = 0; sl < 3; sl++) {
    bool act = (lane + 32 * sl) < 80;
    e0[sl] = act ? __expf(l0[sl] - m0) : 0.f;
    e1[sl] = act ? __expf(l1[sl] - m1) : 0.f;
    s0 += e0[sl]; s1 += e1[sl];
  }
  for (int off = 16; off; off >>= 1) {
    s0 += __shfl_xor(s0, off, 32);
    s1 += __shfl_xor(s1, off, 32);
  }
  float inv0 = 1.f / s0, inv1 = 1.f / s1;
  float a0[16], a1[16];
#pragma unroll
  for (int d = 0; d < 16; d++) { a0[d] = 0.f; a1[d] = 0.f; }
#pragma unroll
  for (int sl = 0; sl < 3; sl++) {
    int kk = lane + 32 * sl;
    if (kk < 80) {
      const float* vp = v_t + ((size_t)b * T_ + tkk[sl]) * C_ + h * HD_;
      float p0 = e0[sl] * inv0, p1 = e1[sl] * inv1;
#pragma unroll
      for (int j = 0; j < 4; j++) {
        float4 vv = *(const float4*)(vp + 4 * j);
        a0[4*j] += p0*vv.x; a0[4*j+1] += p0*vv.y; a0[4*j+2] += p0*vv.z; a0[4*j+3] += p0*vv.w;
        a1[4*j] += p1*vv.x; a1[4*j+1] += p1*vv.y; a1[4*j+2] += p1*vv.z; a1[4*j+3] += p1*vv.w;
      }
    }
  }
  for (int off = 16; off; off >>= 1) {
#pragma unroll
    for (int d = 0; d < 16; d++) {
      a0[d] += __shfl_xor(a0[d], off, 32);
      a1[d] += __shfl_xor(a1[d], off, 32);
    }
  }
  int d = lane & 15;
  float outv = 0.f;
#pragma unroll
  for (int j = 0; j < 16; j++)
    if (d == j) outv = (lane < 16) ? a0[j] : a1[j];
  o_t[((size_t)b * T_ + 2 * r + (lane >> 4)) * C_ + h * HD_ + d] = outv;
}

// ---------------------------------------------------------------------------
// K6: z = o + depthwise-LEPE(v) + lepe_b;  x2 = x + out_w @ z + out_b
// (out-projection via fp32 WMMA). Stores x2 in (B,C,T).
// ---------------------------------------------------------------------------
__global__ __launch_bounds__(256) void k_out(
    const float* __restrict__ o_t, const float* __restrict__ v_t,
    const float* __restrict__ lw, const float* __restrict__ lb,
    const float* __restrict__ ow, const float* __restrict__ ob,
    const float* __restrict__ x, float* __restrict__ x2ct) {
  __shared__ float zs[64][32];
  int bid = blockIdx.x;
  int b = bid >> 6;
  int t0 = (bid & 63) << 5;
  for (int i = threadIdx.x; i < 64 * 32; i += 256) {
    int tt = i >> 6, c = i & 63;
    int t = t0 + tt;
    float vm = (t > 0)      ? v_t[((size_t)b*T_ + t - 1)*C_ + c] : 0.f;
    float v0 = v_t[((size_t)b*T_ + t)*C_ + c];
    float vp = (t < T_ - 1) ? v_t[((size_t)b*T_ + t + 1)*C_ + c] : 0.f;
    float lep = vm * lw[c*3] + v0 * lw[c*3+1] + vp * lw[c*3+2];
    zs[c][tt] = o_t[((size_t)b*T_ + t)*C_ + c] + lep + lb[c];
  }
  __syncthreads();
  int wave = threadIdx.x >> 5, lane = threadIdx.x & 31;
  int row = lane & 15, kh = lane >> 4;
  int co0 = (wave & 3) << 4, tn0 = (wave >> 2) << 4;
  v8f acc = {};
  for (int c0 = 0; c0 < 64; c0 += 4) {
    int kA = c0 + (kh << 1);
    v2f a, bvec;
    a.x = ow[(co0 + row) * 64 + kA];
    a.y = ow[(co0 + row) * 64 + kA + 1];
    bvec.x = zs[kA][tn0 + row];
    bvec.y = zs[kA + 1][tn0 + row];
    acc = wmma4(a, bvec, acc);
  }
#pragma unroll
  for (int v = 0; v < 8; v++) {
    int co = co0 + v + (kh << 3);
    int t  = t0 + tn0 + row;
    size_t o = ((size_t)b * C_ + co) * T_ + t;
    x2ct[o] = acc[v] + ob[co] + x[o];
  }
}

// ---------------------------------------------------------------------------
// K7: fc1 up-proj 64->128 with n2 pre-norm and bn1+relu post (fp32 WMMA).
// Output h1 layout (B, C2, T) so FFT reads contiguous tokens.
// ---------------------------------------------------------------------------
__global__ __launch_bounds__(256) void k_fc1(
    const float* __restrict__ x2ct, const float* __restrict__ n2g,
    const float* __restrict__ n2b, const float* __restrict__ f1w,
    const float* __restrict__ b1g, const float* __restrict__ b1b,
    float* __restrict__ h1) {
  __shared__ float ys[64][32];
  int bid = blockIdx.x;
  int b = bid >> 6;
  int t0 = (bid & 63) << 5;
  for (int i = threadIdx.x; i < 64 * 32; i += 256) {
    int c = i >> 5, tt = i & 31;
    ys[c][tt] = x2ct[((size_t)b*C_ + c)*T_ + t0 + tt] * (n2g[c] * RSQ) + n2b[c];
  }
  __syncthreads();
  int wave = threadIdx.x >> 5, lane = threadIdx.x & 31;
  int row = lane & 15, kh = lane >> 4;
  int cc0 = wave << 4;
  for (int nt = 0; nt < 2; nt++) {
    int tn0 = nt << 4;
    v8f acc = {};
    for (int c0 = 0; c0 < 64; c0 += 4) {
      int kA = c0 + (kh << 1);
      v2f a, bvec;
      a.x = f1w[(cc0 + row) * 64 + kA];
      a.y = f1w[(cc0 + row) * 64 + kA + 1];
      bvec.x = ys[kA][tn0 + row];
      bvec.y = ys[kA + 1][tn0 + row];
      acc = wmma4(a, bvec, acc);
    }
#pragma unroll
    for (int v = 0; v < 8; v++) {
      int cc = cc0 + v + (kh << 3);
      int t  = t0 + tn0 + row;
      float hh = acc[v] * (b1g[cc] * RSQ) + b1b[cc];
      h1[((size_t)b * C2_ + cc) * T_ + t] = fmaxf(hh, 0.f);
    }
  }
}

// ---------------------------------------------------------------------------
// K8: per-(b,cc) token FFT (Stockham radix-2, 2048 pts, LDS ping-pong),
// diagonal complex mix + relu (einsum('bnc,cc->bnc') == diag scaling),
// inverse FFT, take real. Ortho norm = 1/sqrt(N) per direction.
// ---------------------------------------------------------------------------
__device__ __forceinline__ void fft2048(float*& xr, float*& xi,
                                        float*& yr, float*& yi, float sign) {
  int s = 1, ls = 0;
  for (int ncur = 2048; ncur > 1; ncur >>= 1) {
    int m = ncur >> 1;
    float th0 = sign * 6.28318530717958647692f / (float)ncur;
    for (int u = threadIdx.x; u < 1024; u += 128) {
      int q = u & (s - 1);
      int p = u >> ls;
      int i0 = q + (p << ls);
      int i1 = i0 + (m << ls);
      float ar = xr[i0], ai = xi[i0];
      float br = xr[i1], bi = xi[i1];
      int j0 = q + (p << (ls + 1));
      yr[j0] = ar + br;
      yi[j0] = ai + bi;
      float dr = ar - br, di = ai - bi;
      float sn, cs;
      __sincosf(th0 * (float)p, &sn, &cs);
      yr[j0 + s] = dr * cs - di * sn;
      yi[j0 + s] = dr * sn + di * cs;
    }
    __syncthreads();
    float* t = xr; xr = yr; yr = t;
    t = xi; xi = yi; yi = t;
    s <<= 1; ++ls;
  }
}

__global__ __launch_bounds__(128) void k_fft(
    const float* __restrict__ h1, const float* __restrict__ fr,
    const float* __restrict__ fi, const float* __restrict__ frb,
    const float* __restrict__ fib, float* __restrict__ g) {
  __shared__ float A_r[2048], A_i[2048], B_r[2048], B_i[2048];
  int bc = blockIdx.x;                              // b*128 + cc
  int cc = bc & 127;
  const float* src = h1 + (size_t)bc * T_;
  for (int i = threadIdx.x; i < 2048; i += 128) { A_r[i] = src[i]; A_i[i] = 0.f; }
  __syncthreads();
  float *xr = A_r, *xi = A_i, *yr = B_r, *yi = B_i;
  fft2048(xr, xi, yr, yi, -1.f);                    // forward
  float dr = fr[cc * 129];                          // fr[cc,cc]
  float di = fi[cc * 129];                          // fi[cc,cc]
  float rb = frb[cc], ib = fib[cc];
  for (int i = threadIdx.x; i < 2048; i += 128) {
    float re = xr[i] * INVS, im = xi[i] * INVS;     // ortho forward norm
    xr[i] = fmaxf(re * dr - im * di + rb, 0.f);
    xi[i] = fmaxf(im * dr + re * di + ib, 0.f);
  }
  __syncthreads();
  fft2048(xr, xi, yr, yi, 1.f);                     // inverse (unnorm sum)
  float* dst = g + (size_t)bc * T_;
  for (int i = threadIdx.x; i < 2048; i += 128)
    dst[i] = xr[i] * INVS;                          // ortho inverse norm, real
}

// ---------------------------------------------------------------------------
// K9: fc2 down-proj 128->64 + bn2 + final residual, write (B,C,T) output.
// ---------------------------------------------------------------------------
__global__ __launch_bounds__(256) void k_fc2(
    const float* __restrict__ g, const float* __restrict__ f2w,
    const float* __restrict__ b2g, const float* __restrict__ b2b,
    const float* __restrict__ x2ct, float* __restrict__ out) {
  __shared__ float gs[128][32];
  int bid = blockIdx.x;
  int b = bid >> 6;
  int t0 = (bid & 63) << 5;
  for (int i = threadIdx.x; i < 128 * 32; i += 256) {
    int cc = i >> 5, tt = i & 31;
    gs[cc][tt] = g[((size_t)b * C2_ + cc) * T_ + t0 + tt];
  }
  __syncthreads();
  int wave = threadIdx.x >> 5, lane = threadIdx.x & 31;
  int row = lane & 15, kh = lane >> 4;
  int co0 = (wave & 3) << 4, tn0 = (wave >> 2) << 4;
  v8f acc = {};
  for (int c0 = 0; c0 < 128; c0 += 4) {
    int kA = c0 + (kh << 1);
    v2f a, bvec;
    a.x = f2w[(co0 + row) * 128 + kA];
    a.y = f2w[(co0 + row) * 128 + kA + 1];
    bvec.x = gs[kA][tn0 + row];
    bvec.y = gs[kA + 1][tn0 + row];
    acc = wmma4(a, bvec, acc);
  }
#pragma unroll
  for (int v = 0; v < 8; v++) {
    int co = co0 + v + (kh << 3);
    int t  = t0 + tn0 + row;
    size_t o = ((size_t)b * C_ + co) * T_ + t;
    out[o] = x2ct[o] + acc[v] * (b2g[co] * RSQ) + b2b[co];
  }
}

// ---------------------------------------------------------------------------
extern "C" void kernel_launch(void* const* d_in, const int* in_sizes, int n_in,
                              void* d_out, int out_size, void* d_ws,
                              size_t ws_size, hipStream_t stream) {
  const float* x   = (const float*)d_in[0];
  const float* n1g = (const float*)d_in[1];
  const float* n1b = (const float*)d_in[2];
  const float* wq  = (const float*)d_in[3];
  const float* bqg = (const float*)d_in[4];
  const float* bqb = (const float*)d_in[5];
  const float* wk  = (const float*)d_in[6];
  const float* bkg = (const float*)d_in[7];
  const float* bkb = (const float*)d_in[8];
  const float* wv  = (const float*)d_in[9];
  const float* lw  = (const float*)d_in[10];
  const float* lb  = (const float*)d_in[11];
  const float* ow  = (const float*)d_in[12];
  const float* ob  = (const float*)d_in[13];
  const float* n2g = (const float*)d_in[14];
  const float* n2b = (const float*)d_in[15];
  const float* f1w = (const float*)d_in[16];
  const float* b1g = (const float*)d_in[17];
  const float* b1b = (const float*)d_in[18];
  const float* fr  = (const float*)d_in[19];
  const float* fi  = (const float*)d_in[20];
  const float* frb = (const float*)d_in[21];
  const float* fib = (const float*)d_in[22];
  const float* f2w = (const float*)d_in[23];
  const float* b2g = (const float*)d_in[24];
  const float* b2b = (const float*)d_in[25];

  float* ws   = (float*)d_ws;
  float* xn   = ws;                     // 2,097,152 f
  float* q_t  = xn  + 2097152;          // (B,T,C)
  float* k_t  = q_t + 2097152;
  float* v_t  = k_t + 2097152;
  float* qr   = v_t + 2097152;          // (B,NR,C) 1,048,576 f
  float* kr   = qr  + 1048576;
  int*   idxb = (int*)(kr + 1048576);   // B*NR*TK ints (padded to 1M)
  float* o_t  = kr  + 2097152;          // (B,T,C)
  float* x2ct = o_t + 2097152;          // (B,C,T)
  float* h1b  = x2ct + 2097152;         // (B,C2,T) 4,194,304 f
  float* gbuf = h1b + 4194304;          // (B,C2,T)
  float* out  = (float*)d_out;

  k_bn1   <<<8192, 256, 0, stream>>>(x, n1g, n1b, xn);
  k_qkv   <<<1024, 256, 0, stream>>>(xn, wq, wk, wv, bqg, bqb, bkg, bkb,
                                     q_t, k_t, v_t);
  k_reduce<<<4096, 256, 0, stream>>>(q_t, k_t, qr, kr);
  k_route <<<1024, 128, 0, stream>>>(qr, kr, idxb);
  k_attn  <<<8192, 256, 0, stream>>>(q_t, k_t, v_t, idxb, o_t);
  k_out   <<<1024, 256, 0, stream>>>(o_t, v_t, lw, lb, ow, ob, x, x2ct);
  k_fc1   <<<1024, 256, 0, stream>>>(x2ct, n2g, n2b, f1w, b1g, b1b, h1b);
  k_fft   <<<2048, 128, 0, stream>>>(h1b, fr, fi, frb, fib, gbuf);
  k_fc2   <<<1024, 256, 0, stream>>>(gbuf, f2w, b2g, b2b, x2ct, out);

  (void)in_sizes; (void)n_in; (void)out_size; (void)ws_size;
}